// model_64123861729933
// MI455X (gfx1250) — compile-verified
//
#include <hip/hip_runtime.h>
#include <stdint.h>

// ---------------------------------------------------------------------------
// CenterNet-style corner decode, MI455X (gfx1250) version.
// 3 streaming passes over the heatmaps (radix-select histogram x2 + compact),
// staged through LDS with a 4-deep global_load_async_to_lds_b128 ring
// (ASYNCcnt pipelining for HBM latency hiding; pass 2/3 mostly hit the
// 192MB L2), then LDS bitonic-sort kernels for exact top-K / decode / merge.
// ---------------------------------------------------------------------------

#define NBATCH 16
#define NBINS  4096
#define NCAP   4096   // candidate capacity per (batch, layer)
#define NDETS  1000

// ---- helpers ----

__device__ __forceinline__ uint32_t flipkey(uint32_t u) {
  // monotonic float -> u32 key (ascending)
  return u ^ ((uint32_t)((int32_t)u >> 31) | 0x80000000u);
}
__device__ __forceinline__ float unflipf(uint32_t k) {
  uint32_t m = (uint32_t)((int32_t)k >> 31);
  return __uint_as_float(k ^ (~m | 0x80000000u));
}

__device__ __forceinline__ void async_b128(uint32_t lds_byte_off, const void* gaddr) {
  // CDNA5 async copy global -> LDS (ASYNCcnt), per-lane 16B
  asm volatile("global_load_async_to_lds_b128 %0, %1, off"
               :: "v"(lds_byte_off), "v"(gaddr) : "memory");
}
template <int N>
__device__ __forceinline__ void wait_async_le() {
  if constexpr (N <= 0)      asm volatile("s_wait_asynccnt 0x0" ::: "memory");
  else if constexpr (N == 1) asm volatile("s_wait_asynccnt 0x1" ::: "memory");
  else if constexpr (N == 2) asm volatile("s_wait_asynccnt 0x2" ::: "memory");
  else                       asm volatile("s_wait_asynccnt 0x3" ::: "memory");
}
__device__ __forceinline__ void wait_ds_0() {
  asm volatile("s_wait_dscnt 0x0" ::: "memory");
}

// ---- 4-deep async-ring streaming over one batch's heat (float4 granules) ---
// Each lane stages its own 16B slice; per-wave in-flight = 4 x 512B.
// In-order ASYNCcnt completion lets us wait with a compile-time immediate.

template <typename Proc>
__device__ __forceinline__ void stream_heat4(const float4* __restrict__ src,
                                             int first4, int stride4, int nIter,
                                             Proc proc) {
  __shared__ float4 stage[4][256];
  const int tid = threadIdx.x;
  uint32_t lofs[4];
#pragma unroll
  for (int s = 0; s < 4; ++s) lofs[s] = (uint32_t)(uintptr_t)&stage[s][tid];

  int issued = 0;
#pragma unroll
  for (int p = 0; p < 3; ++p)
    if (p < nIter) { async_b128(lofs[p], src + first4 + p * stride4); ++issued; }

  for (int it = 0; it < nIter; ++it) {
    wait_ds_0();                         // prior ds_load of reused stage retired
    if (issued < nIter) {
      async_b128(lofs[issued & 3], src + first4 + issued * stride4);
      ++issued;
    }
    const int pending = issued - it - 1; // allowed outstanding after our slot lands
    if      (pending >= 3) wait_async_le<3>();
    else if (pending == 2) wait_async_le<2>();
    else if (pending == 1) wait_async_le<1>();
    else                   wait_async_le<0>();
    float4 v = stage[it & 3][tid];
    proc(v, first4 + it * stride4);
  }
}

// descending bitonic sort on LDS (key,payload); N power of two
__device__ void bitonic_desc(uint32_t* key, uint32_t* pay, int N, int nthr) {
  const int tid = threadIdx.x;
  for (int k = 2; k <= N; k <<= 1) {
    for (int j = k >> 1; j > 0; j >>= 1) {
      __syncthreads();
      for (int i = tid; i < N; i += nthr) {
        int ixj = i ^ j;
        if (ixj > i) {
          uint32_t a = key[i], b = key[ixj];
          bool up = ((i & k) == 0);
          if ((a < b) == up) {
            key[i] = b; key[ixj] = a;
            uint32_t t = pay[i]; pay[i] = pay[ixj]; pay[ixj] = t;
          }
        }
      }
    }
  }
  __syncthreads();
}

// ---- pass 0: zero workspace control region ----

__global__ __launch_bounds__(256) void zero_kernel(uint32_t* p, int n) {
  for (int i = blockIdx.x * 256 + threadIdx.x; i < n; i += gridDim.x * 256) p[i] = 0;
}

// ---- streaming histogram (level 0: key[31:20]; level 1: refine inside bin t1) ----

__global__ __launch_bounds__(256) void hist_kernel(const float* __restrict__ heat,
                                                   int count4pb,                 // float4 per batch
                                                   uint32_t* __restrict__ histL, // [NBATCH][NBINS]
                                                   const uint32_t* __restrict__ t1L,
                                                   int level) {
  __shared__ uint32_t lhist[NBINS];
  const int tid = threadIdx.x;
  const int b = blockIdx.y;
  for (int i = tid; i < NBINS; i += 256) lhist[i] = 0;
  __syncthreads();

  const uint32_t t1v = (level == 1) ? t1L[b] : 0u;
  const float4* src = ((const float4*)heat) + (size_t)b * count4pb;
  const int stride4 = gridDim.x * 256;
  const int nIter = count4pb / stride4;  // launch geometry chosen so this is exact

  stream_heat4(src, blockIdx.x * 256 + tid, stride4, nIter,
    [&](float4 v, int /*idx4*/) {
      uint32_t u[4] = {__float_as_uint(v.x), __float_as_uint(v.y),
                       __float_as_uint(v.z), __float_as_uint(v.w)};
#pragma unroll
      for (int c = 0; c < 4; ++c) {
        uint32_t key = flipkey(u[c]);
        if (level == 0) {
          atomicAdd(&lhist[key >> 20], 1u);
        } else if ((key >> 20) == t1v) {
          atomicAdd(&lhist[(key >> 8) & 0xFFFu], 1u);
        }
      }
    });

  __syncthreads();
  for (int i = tid; i < NBINS; i += 256) {
    uint32_t c = lhist[i];
    if (c) atomicAdd(&histL[(size_t)b * NBINS + i], c);
  }
}

// ---- descending scan over 4096 bins: find cut bin / final threshold ----

__global__ __launch_bounds__(256) void scan_kernel(const uint32_t* __restrict__ hist,
                                                   uint32_t* __restrict__ t1,
                                                   uint32_t* __restrict__ rem,
                                                   uint32_t* __restrict__ thr,
                                                   int level) {
  const int b = blockIdx.x, l = blockIdx.y;
  const int lb = l * NBATCH + b;
  const uint32_t* h = hist + (size_t)lb * NBINS;
  __shared__ uint32_t part[256];
  __shared__ uint32_t pref[256];
  const int tid = threadIdx.x;
  const int hi = 4095 - 16 * tid;
  uint32_t s = 0;
#pragma unroll
  for (int q = 0; q < 16; ++q) s += h[hi - q];
  part[tid] = s;
  __syncthreads();
  if (tid == 0) {
    uint32_t run = 0;
    for (int t = 0; t < 256; ++t) { pref[t] = run; run += part[t]; }
  }
  __syncthreads();
  const uint32_t KL[3] = {2000u, 2000u, 1024u};   // K = min(2000, H*W)
  const uint32_t target = (level == 0) ? KL[l] : rem[lb];
  uint32_t cum = pref[tid];
  if (target > 0 && cum < target && cum + part[tid] >= target) {
    for (int q = 0; q < 16; ++q) {
      uint32_t c = h[hi - q];
      if (cum + c >= target) {
        uint32_t T = (uint32_t)(hi - q);
        if (level == 0) { t1[lb] = T; rem[lb] = target - cum; }
        else            { thr[lb] = (t1[lb] << 20) | (T << 8); }
        break;
      }
      cum += c;
    }
  }
}

// ---- streaming compaction of candidates >= threshold ----

__global__ __launch_bounds__(256) void compact_kernel(const float* __restrict__ heat,
                                                      int count4pb,
                                                      const uint32_t* __restrict__ thrL,
                                                      uint32_t* __restrict__ cntL,
                                                      uint2* __restrict__ candL) {
  const int tid = threadIdx.x;
  const int b = blockIdx.y;
  const uint32_t thrv = thrL[b];
  uint32_t* cntb = cntL + b;
  uint2* candb = candL + (size_t)b * NCAP;
  const float4* src = ((const float4*)heat) + (size_t)b * count4pb;
  const int stride4 = gridDim.x * 256;
  const int nIter = count4pb / stride4;

  stream_heat4(src, blockIdx.x * 256 + tid, stride4, nIter,
    [&](float4 v, int idx4) {
      uint32_t u[4] = {__float_as_uint(v.x), __float_as_uint(v.y),
                       __float_as_uint(v.z), __float_as_uint(v.w)};
#pragma unroll
      for (int c = 0; c < 4; ++c) {
        uint32_t key = flipkey(u[c]);
        if (key >= thrv) {
          uint32_t pos = atomicAdd(cntb, 1u);
          if (pos < NCAP) candb[pos] = make_uint2(key, (uint32_t)(idx4 * 4 + c));
        }
      }
    });
}

// ---- per-(batch,layer): sort candidates, decode boxes, per-layer top-1000 ----

__global__ __launch_bounds__(512) void finalize_kernel(
    const float* __restrict__ tl0, const float* __restrict__ tl1, const float* __restrict__ tl2,
    const float* __restrict__ br0, const float* __restrict__ br1, const float* __restrict__ br2,
    const uint2* __restrict__ cand, const uint32_t* __restrict__ cnt,
    float* __restrict__ det) {
  const int b = blockIdx.x, l = blockIdx.y;
  const int lb = l * NBATCH + b;
  const int tid = threadIdx.x;
  const int HW = (l == 0) ? 16384 : (l == 1) ? 4096 : 1024;
  const int W  = (l == 0) ? 128   : (l == 1) ? 64   : 32;
  const int K  = (l == 2) ? 1024 : 2000;
  const float scale = (l == 0) ? 8.f : (l == 1) ? 16.f : 32.f;
  const float* tl = (l == 0) ? tl0 : (l == 1) ? tl1 : tl2;
  const float* br = (l == 0) ? br0 : (l == 1) ? br1 : br2;

  __shared__ uint32_t key[NCAP];
  __shared__ uint32_t pay[NCAP];
  __shared__ uint32_t key2[2048];
  __shared__ uint32_t pay2[2048];
  __shared__ float bx[2048][4];

  uint32_t n = cnt[lb]; if (n > NCAP) n = NCAP;
  const uint2* cb = cand + (size_t)lb * NCAP;
  for (int i = tid; i < NCAP; i += 512) {
    if ((uint32_t)i < n) { uint2 c = cb[i]; key[i] = c.x; pay[i] = c.y; }
    else                 { key[i] = 0u; pay[i] = 0u; }
  }
  bitonic_desc(key, pay, NCAP, 512);     // exact top-K now at the front

  const float* tlb = tl + (size_t)b * 2 * HW;
  const float* brb = br + (size_t)b * 2 * HW;
  for (int i = tid; i < 2048; i += 512) {
    if (i < K) {
      float score = unflipf(key[i]);
      uint32_t idx = pay[i];
      uint32_t sp = idx % (uint32_t)HW;
      float xs = (float)(sp % (uint32_t)W);
      float ys = (float)(sp / (uint32_t)W);
      float t0 = tlb[sp], t1v = tlb[HW + sp];
      float b0 = brb[sp], b1v = brb[HW + sp];
      float tlx = xs - (1.5f * t0 + 2.25f);
      float tly = ys - (1.5f * t1v + 2.25f);
      float brx = xs + (1.5f * b0 + 2.25f);
      float bry = ys + (1.5f * b1v + 2.25f);
      if (brx < tlx || bry < tly) score = -1.0f;
      bx[i][0] = tlx; bx[i][1] = tly; bx[i][2] = brx; bx[i][3] = bry;
      key2[i] = flipkey(__float_as_uint(score));
      pay2[i] = (uint32_t)i;
    } else {
      key2[i] = 0u; pay2[i] = 0u;
    }
  }
  bitonic_desc(key2, pay2, 2048, 512);   // re-rank after invalidation

  for (int i = tid; i < NDETS; i += 512) {
    float* d = det + ((size_t)lb * NDETS + i) * 7;
    if (i == 6) {                         // reference quirk: row 6 := layer idx
      for (int c = 0; c < 7; ++c) d[c] = (float)l;
    } else {
      float sc = unflipf(key2[i]);
      uint32_t s = pay2[i];
      d[0] = sc;
      d[1] = bx[s][0] * scale; d[2] = bx[s][1] * scale;
      d[3] = bx[s][2] * scale; d[4] = bx[s][3] * scale;
      d[5] = 0.f; d[6] = 0.f;
    }
  }
}

// ---- per-batch merge: top-1000 of 3x1000 ----

__global__ __launch_bounds__(512) void merge_kernel(const float* __restrict__ det,
                                                    float* __restrict__ out) {
  const int b = blockIdx.x;
  const int tid = threadIdx.x;
  __shared__ uint32_t key[4096];
  __shared__ uint32_t pay[4096];
  for (int i = tid; i < 4096; i += 512) {
    if (i < 3 * NDETS) {
      int l = i / NDETS, j = i % NDETS;
      float s = det[((size_t)(l * NBATCH + b) * NDETS + j) * 7];
      key[i] = flipkey(__float_as_uint(s));
      pay[i] = (uint32_t)i;
    } else { key[i] = 0u; pay[i] = 0u; }
  }
  bitonic_desc(key, pay, 4096, 512);
  for (int i = tid; i < NDETS; i += 512) {
    uint32_t s = pay[i];
    int l = (int)(s / NDETS), j = (int)(s % NDETS);
    const float* src = det + ((size_t)(l * NBATCH + b) * NDETS + j) * 7;
    float* o = out + ((size_t)b * NDETS + i) * 7;
    for (int c = 0; c < 7; ++c) o[c] = src[c];
  }
}

// ---------------------------------------------------------------------------

extern "C" void kernel_launch(void* const* d_in, const int* in_sizes, int n_in,
                              void* d_out, int out_size, void* d_ws, size_t ws_size,
                              hipStream_t stream) {
  (void)in_sizes; (void)n_in; (void)out_size; (void)ws_size;
  const float* heat[3] = {(const float*)d_in[0], (const float*)d_in[3], (const float*)d_in[6]};
  const float* tlp[3]  = {(const float*)d_in[1], (const float*)d_in[4], (const float*)d_in[7]};
  const float* brp[3]  = {(const float*)d_in[2], (const float*)d_in[5], (const float*)d_in[8]};

  uint8_t* ws = (uint8_t*)d_ws;
  uint32_t* histA = (uint32_t*)(ws + 0);          // 3*16*4096 u32
  uint32_t* histB = (uint32_t*)(ws + 786432);     // 3*16*4096 u32
  uint32_t* t1    = (uint32_t*)(ws + 1572864);    // 48 u32
  uint32_t* rem   = (uint32_t*)(ws + 1573056);
  uint32_t* thr   = (uint32_t*)(ws + 1573248);
  uint32_t* cnt   = (uint32_t*)(ws + 1573440);
  uint2*    cand  = (uint2*)   (ws + 1573632);    // 48*4096 uint2
  float*    det   = (float*)   (ws + 3146496);    // 3*16*1000*7 f32

  // zero hists + control (393408 u32 words)
  zero_kernel<<<512, 256, 0, stream>>>((uint32_t*)d_ws, 393408);

  const int count4[3] = {327680, 81920, 20480};   // C*H*W/4 per batch
  const int chunks[3] = {64, 32, 16};             // exact-tiling grid.x per layer

  // pass 1: coarse histogram (key bits [31:20])
  for (int l = 0; l < 3; ++l)
    hist_kernel<<<dim3(chunks[l], NBATCH), 256, 0, stream>>>(
        heat[l], count4[l], histA + (size_t)l * NBATCH * NBINS, t1 + l * NBATCH, 0);
  scan_kernel<<<dim3(NBATCH, 3), 256, 0, stream>>>(histA, t1, rem, thr, 0);

  // pass 2: refine inside cut bin (key bits [19:8]) — mostly L2 hits (192MB L2)
  for (int l = 0; l < 3; ++l)
    hist_kernel<<<dim3(chunks[l], NBATCH), 256, 0, stream>>>(
        heat[l], count4[l], histB + (size_t)l * NBATCH * NBINS, t1 + l * NBATCH, 1);
  scan_kernel<<<dim3(NBATCH, 3), 256, 0, stream>>>(histB, t1, rem, thr, 1);

  // pass 3: compact candidates >= threshold — mostly L2 hits
  for (int l = 0; l < 3; ++l)
    compact_kernel<<<dim3(chunks[l], NBATCH), 256, 0, stream>>>(
        heat[l], count4[l], thr + l * NBATCH, cnt + l * NBATCH,
        cand + (size_t)l * NBATCH * NCAP);

  // per-(b,l) decode + per-layer top-1000, then per-batch merge
  finalize_kernel<<<dim3(NBATCH, 3), 512, 0, stream>>>(
      tlp[0], tlp[1], tlp[2], brp[0], brp[1], brp[2], cand, cnt, det);
  merge_kernel<<<NBATCH, 512, 0, stream>>>(det, (float*)d_out);
}